// Saint_84086869721202
// MI455X (gfx1250) — compile-verified
//
#include <hip/hip_runtime.h>

#define N_NODES 20000
#define N_EDGES 320000
#define NUM_FEAT 14
#define DIM 256

typedef float v2f __attribute__((ext_vector_type(2)));
typedef float v4f __attribute__((ext_vector_type(4)));
typedef float v8f __attribute__((ext_vector_type(8)));

// ---------------- utility kernels ----------------

__global__ void k_zero(float* __restrict__ p, int n) {
    int i = blockIdx.x * blockDim.x + threadIdx.x;
    if (i < n) p[i] = 0.0f;
}

__global__ void k_count_deg(const long long* __restrict__ dst, float* __restrict__ cnt, int e) {
    int i = blockIdx.x * blockDim.x + threadIdx.x;
    if (i < e) atomicAdd(&cnt[(int)dst[i]], 1.0f);
}

__global__ void k_invert(float* __restrict__ cnt, int n) {
    int i = blockIdx.x * blockDim.x + threadIdx.x;
    if (i < n) cnt[i] = 1.0f / fmaxf(cnt[i], 1.0f);
}

// agg[n, f] *= cnt_inv[n]   (F = 1<<shift; turns sum-agg into mean-agg once,
// instead of re-scaling inside the GEMM inner loop 16x per element)
__global__ void k_scale_rows(float* __restrict__ agg, const float* __restrict__ cnt_inv,
                             int shift, int total) {
    int i = blockIdx.x * blockDim.x + threadIdx.x;
    if (i < total) agg[i] *= cnt_inv[i >> shift];
}

// pad x [N,14] -> xp [N,16] (zero cols 14,15)
__global__ void k_pad_x(const float* __restrict__ x, float* __restrict__ xp, int n) {
    int i = blockIdx.x * blockDim.x + threadIdx.x;   // N*16 total
    if (i < n) {
        int node = i >> 4, f = i & 15;
        xp[i] = (f < NUM_FEAT) ? x[node * NUM_FEAT + f] : 0.0f;
    }
}

// repack row-major W[K,256] into K4-packed Bp so each lane's (k, k+1) B pair is
// one contiguous b64:  Bp[((k>>2)*256 + col)*4 + (k&3)] = W[k*256 + col]
__global__ void k_repack_w(const float* __restrict__ w, float* __restrict__ bp, int total) {
    int i = blockIdx.x * blockDim.x + threadIdx.x;   // K*256 total
    if (i < total) {
        int k = i >> 8, col = i & 255;
        bp[((size_t)(k >> 2) * DIM + col) * 4 + (k & 3)] = w[i];
    }
}

// pad W[14,256] -> K=16 and repack in one pass
__global__ void k_pad_repack_w(const float* __restrict__ w, float* __restrict__ bp) {
    int i = blockIdx.x * blockDim.x + threadIdx.x;   // 16*256 total
    if (i < 16 * DIM) {
        int k = i >> 8, col = i & 255;
        float v = (k < NUM_FEAT) ? w[(size_t)k * DIM + col] : 0.0f;
        bp[((size_t)(k >> 2) * DIM + col) * 4 + (k & 3)] = v;
    }
}

// ---------------- scatter-mean message passing ----------------

// F=256: one wave per edge. Indices/weight are wave-uniform loads; features move
// as two b128 loads per lane (8 floats/lane). 8 waves (8 edges) per block.
__global__ void k_scatter256(const float* __restrict__ x,
                             const long long* __restrict__ src,
                             const long long* __restrict__ dst,
                             const float* __restrict__ ew,
                             float* __restrict__ agg) {
    const int wave = threadIdx.x >> 5;
    const int lane = threadIdx.x & 31;
    const int e = blockIdx.x * 8 + wave;
    const int s = (int)src[e];
    const int d = (int)dst[e];
    const float w = ew[e];
    const float* xs = x + (size_t)s * DIM + lane * 8;
    float* ad = agg + (size_t)d * DIM + lane * 8;
    v4f a0 = *(const v4f*)(xs);
    v4f a1 = *(const v4f*)(xs + 4);
#pragma unroll
    for (int j = 0; j < 4; ++j) atomicAdd(&ad[j], a0[j] * w);
#pragma unroll
    for (int j = 0; j < 4; ++j) atomicAdd(&ad[4 + j], a1[j] * w);
}

// F=16 (padded layer 1): half-wave per edge, one feature per lane.
__global__ void k_scatter16(const float* __restrict__ xp,
                            const long long* __restrict__ src,
                            const long long* __restrict__ dst,
                            const float* __restrict__ ew,
                            float* __restrict__ agg) {
    const int e = blockIdx.x * 16 + (threadIdx.x >> 4);
    const int f = threadIdx.x & 15;
    const int s = (int)src[e];
    const int d = (int)dst[e];
    const float w = ew[e];
    atomicAdd(&agg[(size_t)d * 16 + f], xp[(size_t)s * 16 + f] * w);
}

// ---------------- fused WMMA GEMM ----------------
// One wave computes T vertically-stacked 16x16 tiles (T*16 rows x 16 cols),
// so every B b64 load feeds T v_wmma_f32_16x16x4_f32 ops. No per-element
// scaling in the loop: loads feed WMMA directly.

template <int T>
__device__ __forceinline__ void wmma_chain(v8f (&c)[T],
                                           const float* __restrict__ A,
                                           const float* __restrict__ Bp,
                                           int K, int lda,
                                           int row0, int lm, int col, int koff) {
    for (int k = 0; k < K; k += 4) {
        v2f b = *(const v2f*)(Bp + ((size_t)(k >> 2) * DIM + col) * 4 + koff);
#pragma unroll
        for (int t = 0; t < T; ++t) {
            const int m = row0 + t * 16 + lm;
            v2f a = *(const v2f*)(A + (size_t)m * lda + k + koff);
            c[t] = __builtin_amdgcn_wmma_f32_16x16x4_f32(false, a, false, b,
                                                         (short)0, c[t], false, false);
        }
    }
}

template <int T>
__global__ void k_gemm_fused(const float* __restrict__ A1, const float* __restrict__ B1,
                             int K1, int lda1,
                             const float* __restrict__ A2, const float* __restrict__ B2,
                             int K2, int lda2,
                             const float* __restrict__ A3, const float* __restrict__ B3,
                             int K3, int lda3,
                             const float* __restrict__ bias,
                             float* __restrict__ out, int do_relu) {
    const int lane = threadIdx.x;          // 0..31 (wave32)
    const int half = lane >> 4;            // K split: lanes 16-31 hold K+2,K+3
    const int lm   = lane & 15;
    const int koff = half * 2;
    const int row0 = blockIdx.x * (16 * T);
    const int col0 = blockIdx.y * 16;
    const int col  = col0 + lm;

    v8f c[T];
#pragma unroll
    for (int t = 0; t < T; ++t) c[t] = (v8f){};

    wmma_chain<T>(c, A1, B1, K1, lda1, row0, lm, col, koff);
    if (K2 > 0)
        wmma_chain<T>(c, A2, B2, K2, lda2, row0, lm, col, koff);
    if (K3 > 0)
        wmma_chain<T>(c, A3, B3, K3, lda3, row0, lm, col, koff);

    // C/D layout: VGPR r -> M = r (lanes 0-15) or r+8 (lanes 16-31), N = lane&15
    const float bn = bias[col];
#pragma unroll
    for (int t = 0; t < T; ++t) {
#pragma unroll
        for (int r = 0; r < 8; ++r) {
            const int mr = row0 + t * 16 + r + half * 8;
            float v = c[t][r] + bn;
            if (do_relu) v = fmaxf(v, 0.0f);
            out[(size_t)mr * DIM + col0 + lm] = v;
        }
    }
}

// ---------------- host orchestration ----------------

static inline int ceil_div(int a, int b) { return (a + b - 1) / b; }

extern "C" void kernel_launch(void* const* d_in, const int* in_sizes, int n_in,
                              void* d_out, int out_size, void* d_ws, size_t ws_size,
                              hipStream_t stream) {
    const float*     x       = (const float*)d_in[0];
    const long long* ei      = (const long long*)d_in[1];   // int64 [2, E]
    const float*     ew      = (const float*)d_in[2];
    const float*     W1_rel  = (const float*)d_in[3];
    const float*     b1_rel  = (const float*)d_in[4];
    const float*     W1_root = (const float*)d_in[5];
    const float*     W2_rel  = (const float*)d_in[6];
    const float*     b2_rel  = (const float*)d_in[7];
    const float*     W2_root = (const float*)d_in[8];
    const float*     W3_rel  = (const float*)d_in[9];
    const float*     b3_rel  = (const float*)d_in[10];
    const float*     W3_root = (const float*)d_in[11];
    const float*     W_lin   = (const float*)d_in[12];
    const float*     b_lin   = (const float*)d_in[13];
    float* out = (float*)d_out;

    const long long* src = ei;
    const long long* dst = ei + N_EDGES;

    // workspace carve-up (floats)
    float* ws      = (float*)d_ws;
    float* cnt_inv = ws;                         // 20000
    float* xp      = cnt_inv + N_NODES;          // 20000*16
    float* bp1r    = xp + (size_t)N_NODES * 16;  // 16*256 repacked
    float* bp1t    = bp1r + 16 * DIM;
    float* bp2r    = bp1t + 16 * DIM;            // 256*256 repacked x7
    float* bp2t    = bp2r + DIM * DIM;
    float* bp3r    = bp2t + DIM * DIM;
    float* bp3t    = bp3r + DIM * DIM;
    float* bpl1    = bp3t + DIM * DIM;
    float* bpl2    = bpl1 + DIM * DIM;
    float* bpl3    = bpl2 + DIM * DIM;
    float* agg     = bpl3 + DIM * DIM;           // 20000*256
    float* x1      = agg + (size_t)N_NODES * DIM;
    float* x2      = x1 + (size_t)N_NODES * DIM;
    float* x3      = x2 + (size_t)N_NODES * DIM;

    constexpr int T = 5;                         // 1250 M-tiles / 5 = 250 blocks
    const dim3 gemm_grid(N_NODES / (16 * T), DIM / 16);
    const dim3 gemm_blk(32);

    // 1) degree -> inverse count
    k_zero<<<ceil_div(N_NODES, 256), 256, 0, stream>>>(cnt_inv, N_NODES);
    k_count_deg<<<ceil_div(N_EDGES, 256), 256, 0, stream>>>(dst, cnt_inv, N_EDGES);
    k_invert<<<ceil_div(N_NODES, 256), 256, 0, stream>>>(cnt_inv, N_NODES);

    // 2) pad layer-1 features, pad+repack layer-1 weights, repack all 256-K weights
    k_pad_x<<<ceil_div(N_NODES * 16, 256), 256, 0, stream>>>(x, xp, N_NODES * 16);
    k_pad_repack_w<<<ceil_div(16 * DIM, 256), 256, 0, stream>>>(W1_rel, bp1r);
    k_pad_repack_w<<<ceil_div(16 * DIM, 256), 256, 0, stream>>>(W1_root, bp1t);
    k_repack_w<<<DIM, 256, 0, stream>>>(W2_rel, bp2r, DIM * DIM);
    k_repack_w<<<DIM, 256, 0, stream>>>(W2_root, bp2t, DIM * DIM);
    k_repack_w<<<DIM, 256, 0, stream>>>(W3_rel, bp3r, DIM * DIM);
    k_repack_w<<<DIM, 256, 0, stream>>>(W3_root, bp3t, DIM * DIM);
    k_repack_w<<<DIM, 256, 0, stream>>>(W_lin,             bpl1, DIM * DIM);
    k_repack_w<<<DIM, 256, 0, stream>>>(W_lin + 256 * DIM, bpl2, DIM * DIM);
    k_repack_w<<<DIM, 256, 0, stream>>>(W_lin + 512 * DIM, bpl3, DIM * DIM);

    // 3) layer 1: aggregate (F=16 padded) -> mean -> fused GEMM + relu -> x1
    k_zero<<<ceil_div(N_NODES * 16, 256), 256, 0, stream>>>(agg, N_NODES * 16);
    k_scatter16<<<N_EDGES / 16, 256, 0, stream>>>(xp, src, dst, ew, agg);
    k_scale_rows<<<ceil_div(N_NODES * 16, 256), 256, 0, stream>>>(
        agg, cnt_inv, 4, N_NODES * 16);
    k_gemm_fused<T><<<gemm_grid, gemm_blk, 0, stream>>>(
        agg, bp1r, 16, 16,
        xp,  bp1t, 16, 16,
        nullptr, nullptr, 0, 0,
        b1_rel, x1, 1);

    // 4) layer 2: aggregate x1 (F=256, wave-per-edge) -> mean -> GEMM + relu -> x2
    k_zero<<<ceil_div(N_NODES * DIM, 256), 256, 0, stream>>>(agg, N_NODES * DIM);
    k_scatter256<<<N_EDGES / 8, 256, 0, stream>>>(x1, src, dst, ew, agg);
    k_scale_rows<<<ceil_div(N_NODES * DIM, 256), 256, 0, stream>>>(
        agg, cnt_inv, 8, N_NODES * DIM);
    k_gemm_fused<T><<<gemm_grid, gemm_blk, 0, stream>>>(
        agg, bp2r, DIM, DIM,
        x1,  bp2t, DIM, DIM,
        nullptr, nullptr, 0, 0,
        b2_rel, x2, 1);

    // 5) layer 3 -> x3
    k_zero<<<ceil_div(N_NODES * DIM, 256), 256, 0, stream>>>(agg, N_NODES * DIM);
    k_scatter256<<<N_EDGES / 8, 256, 0, stream>>>(x2, src, dst, ew, agg);
    k_scale_rows<<<ceil_div(N_NODES * DIM, 256), 256, 0, stream>>>(
        agg, cnt_inv, 8, N_NODES * DIM);
    k_gemm_fused<T><<<gemm_grid, gemm_blk, 0, stream>>>(
        agg, bp3r, DIM, DIM,
        x2,  bp3t, DIM, DIM,
        nullptr, nullptr, 0, 0,
        b3_rel, x3, 1);

    // 6) final linear (concat-free 3-way GEMM):
    //    out = x1 @ W_lin[0:256] + x2 @ W_lin[256:512] + x3 @ W_lin[512:768] + b_lin
    k_gemm_fused<T><<<gemm_grid, gemm_blk, 0, stream>>>(
        x1, bpl1, DIM, DIM,
        x2, bpl2, DIM, DIM,
        x3, bpl3, DIM, DIM,
        b_lin, out, 0);
}